// Recurrent_layer_GCN_73203422593433
// MI455X (gfx1250) — compile-verified
//
#include <hip/hip_runtime.h>
#include <math.h>

// ---------------------------------------------------------------------------
// Recurrent GCN (GCNConv x2 + GRU, 3 lookback steps) for MI455X (gfx1250).
// fp32 WMMA (v_wmma_f32_16x16x4_f32) for every dense matmul; atomic-f32
// scatter for the edge aggregation (L2-resident, memory-bound part).
// ---------------------------------------------------------------------------

typedef __attribute__((ext_vector_type(2))) float v2f;
typedef __attribute__((ext_vector_type(8))) float v8f;

#define HID 128
#define GATE3 384

__device__ __forceinline__ v8f wmma_f32_k4(v2f a, v2f b, v8f c) {
  // D = A(16x4,f32) * B(4x16,f32) + C(16x16,f32)
  return __builtin_amdgcn_wmma_f32_16x16x4_f32(
      /*neg_a=*/false, a, /*neg_b=*/false, b,
      /*c_mod=*/(short)0, c, /*reuse_a=*/false, /*reuse_b=*/false);
}

// ----------------------------- small utility kernels -----------------------

__global__ void k_fill1(float* __restrict__ p, int n) {
  int i = blockIdx.x * blockDim.x + threadIdx.x;
  if (i < n) p[i] = 1.0f;  // self-loop contributes degree 1 to every node
}

__global__ void k_edge_deg(float* __restrict__ deg, const int* __restrict__ dst, int E) {
  int e = blockIdx.x * blockDim.x + threadIdx.x;
  if (e < E) atomicAdd(&deg[dst[e]], 1.0f);
}

__global__ void k_rsqrt(float* __restrict__ d, int n) {
  int i = blockIdx.x * blockDim.x + threadIdx.x;
  if (i < n) d[i] = rsqrtf(d[i]);  // deg >= 1 always (self-loops)
}

__global__ void k_init_h(float* __restrict__ h, const float* __restrict__ hc, int total) {
  int i = blockIdx.x * blockDim.x + threadIdx.x;
  if (i < total) h[i] = hc[i & (HID - 1)];
}

// ----------------------------- generic WMMA GEMM ---------------------------
// C[N x 128] = op(A)[N x K] * B[K x 128] (+ bias), op = optional ReLU.
// Block: 256 threads = 8 waves; 16 output rows per block, one 16-col tile
// per wave. A tile staged through LDS; B (weights) stream from L2.

__global__ __launch_bounds__(256) void k_gemm16(
    const float* __restrict__ A, const float* __restrict__ B,
    const float* __restrict__ bias, float* __restrict__ C,
    int Nrows, int K, int relu_a) {
  __shared__ float sA[16 * HID];
  const int row0 = blockIdx.x * 16;
  const int tid  = threadIdx.x;

  for (int i = tid; i < 16 * K; i += 256) {
    int r = i / K, c = i - r * K;
    int gr = row0 + r;
    float v = (gr < Nrows) ? A[(size_t)gr * K + c] : 0.0f;
    if (relu_a) v = fmaxf(v, 0.0f);
    sA[r * HID + c] = v;
  }
  __syncthreads();

  const int wave = tid >> 5;          // 0..7 : 16-column tile
  const int lane = tid & 31;
  const int m    = lane & 15;         // row within tile (A) / col (B,C)
  const int kb   = (lane >> 4) << 1;  // 0 or 2 (K sub-offset, ISA f32 layout)
  const int mb   = (lane >> 4) << 3;  // 0 or 8 (C row base)
  const int col  = wave * 16 + m;     // output column 0..127

  v8f acc = {};
  for (int k0 = 0; k0 < K; k0 += 4) {
    v2f a, b;
    a.x = sA[m * HID + k0 + kb];
    a.y = sA[m * HID + k0 + kb + 1];
    b.x = B[(size_t)(k0 + kb) * HID + col];
    b.y = B[(size_t)(k0 + kb + 1) * HID + col];
    acc = wmma_f32_k4(a, b, acc);
  }

  const float bv = bias ? bias[col] : 0.0f;
#pragma unroll
  for (int i = 0; i < 8; ++i) {
    int gr = row0 + mb + i;
    if (gr < Nrows) C[(size_t)gr * HID + col] = acc[i] + bv;
  }
}

// ----------------------------- GCN aggregation -----------------------------
// acc[n,:] = bias + hW[n,:] * dis[n]^2        (self-loop term, coalesced)
__global__ void k_scatter_init(float* __restrict__ acc, const float* __restrict__ hW,
                               const float* __restrict__ dis,
                               const float* __restrict__ bias, int total) {
  int i = blockIdx.x * blockDim.x + threadIdx.x;
  if (i >= total) return;
  int n = i >> 7, c = i & (HID - 1);
  float w = dis[n];
  acc[i] = bias[c] + hW[i] * w * w;
}

// acc[dst,:] += hW[src,:] * dis[src]*dis[dst]   (one wave per edge, float4/lane)
__global__ void k_edge_scatter(float* __restrict__ acc, const float* __restrict__ hW,
                               const float* __restrict__ dis,
                               const int* __restrict__ src, const int* __restrict__ dst,
                               int E) {
  int gtid = blockIdx.x * blockDim.x + threadIdx.x;
  int e    = gtid >> 5;
  int lane = gtid & 31;
  if (e >= E) return;
  int s = src[e], d = dst[e];
  float w = dis[s] * dis[d];
  const float4 v = ((const float4*)(hW + (size_t)s * HID))[lane];
  float* ap = acc + (size_t)d * HID + lane * 4;
  atomicAdd(ap + 0, v.x * w);
  atomicAdd(ap + 1, v.y * w);
  atomicAdd(ap + 2, v.z * w);
  atomicAdd(ap + 3, v.w * w);
}

// ----------------------------- fused GRU cell ------------------------------
// 16 nodes per block. gi = xin @ Wih^T, gh = h @ Whh^T via WMMA into LDS,
// then elementwise gate math, h updated in place (rows are block-local).

__global__ __launch_bounds__(256) void k_gru(
    const float* __restrict__ xin, int relu_x, float* __restrict__ h,
    const float* __restrict__ Wih, const float* __restrict__ Whh,
    const float* __restrict__ bih, const float* __restrict__ bhh, int Nrows) {
  __shared__ float sX[16 * HID];
  __shared__ float sH[16 * HID];
  __shared__ float sGi[16 * GATE3];
  __shared__ float sGh[16 * GATE3];
  const int row0 = blockIdx.x * 16;
  const int tid  = threadIdx.x;

  for (int i = tid; i < 16 * HID; i += 256) {
    int r = i >> 7, c = i & (HID - 1);
    int gr = row0 + r;
    float xv = 0.0f, hv = 0.0f;
    if (gr < Nrows) {
      xv = xin[(size_t)gr * HID + c];
      hv = h[(size_t)gr * HID + c];
    }
    if (relu_x) xv = fmaxf(xv, 0.0f);
    sX[i] = xv;
    sH[i] = hv;
  }
  __syncthreads();

  const int wave = tid >> 5;
  const int lane = tid & 31;
  const int m  = lane & 15;
  const int kb = (lane >> 4) << 1;
  const int mb = (lane >> 4) << 3;

  // 48 output tiles: 24 for gi (Wih), 24 for gh (Whh); wave-uniform split.
  for (int t = wave; t < 48; t += 8) {
    const int isI = (t < 24) ? 1 : 0;
    const int ct  = isI ? t : (t - 24);
    const float* __restrict__ W = isI ? Wih : Whh;   // [384 x 128] row-major
    const float* As = isI ? sX : sH;
    float*       G  = isI ? sGi : sGh;
    const int o = ct * 16 + m;                       // gate column 0..383

    v8f acc = {};
    for (int k0 = 0; k0 < HID; k0 += 4) {
      v2f a, b;
      a.x = As[m * HID + k0 + kb];
      a.y = As[m * HID + k0 + kb + 1];
      // gi[n,o] = sum_k x[n,k] * Wih[o,k]  ->  B[k][o] = W[o*128 + k]
      b.x = W[(size_t)o * HID + k0 + kb];
      b.y = W[(size_t)o * HID + k0 + kb + 1];
      acc = wmma_f32_k4(a, b, acc);
    }
#pragma unroll
    for (int i = 0; i < 8; ++i) G[(mb + i) * GATE3 + o] = acc[i];
  }
  __syncthreads();

  for (int i = tid; i < 16 * HID; i += 256) {
    int r = i >> 7, c = i & (HID - 1);
    int gr = row0 + r;
    if (gr >= Nrows) continue;
    float ir = sGi[r * GATE3 + c]           + bih[c];
    float iz = sGi[r * GATE3 + c + HID]     + bih[c + HID];
    float in_= sGi[r * GATE3 + c + 2 * HID] + bih[c + 2 * HID];
    float hr = sGh[r * GATE3 + c]           + bhh[c];
    float hz = sGh[r * GATE3 + c + HID]     + bhh[c + HID];
    float hn = sGh[r * GATE3 + c + 2 * HID] + bhh[c + 2 * HID];
    float rg = 1.0f / (1.0f + __expf(-(ir + hr)));
    float zg = 1.0f / (1.0f + __expf(-(iz + hz)));
    float ng = tanhf(in_ + rg * hn);
    h[(size_t)gr * HID + c] = (1.0f - zg) * ng + zg * sH[i];
  }
}

// ----------------------------- output head --------------------------------

__global__ void k_out_softmax(const float* __restrict__ h, const float* __restrict__ W,
                              const float* __restrict__ b, float* __restrict__ out, int N) {
  int n = blockIdx.x * blockDim.x + threadIdx.x;
  if (n >= N) return;
  float lg[5];
#pragma unroll
  for (int o = 0; o < 5; ++o) lg[o] = b[o];
  const float* hr = h + (size_t)n * HID;
  for (int k = 0; k < HID; ++k) {
    float hv = hr[k];
#pragma unroll
    for (int o = 0; o < 5; ++o) lg[o] += hv * W[k * 5 + o];
  }
  float mx = lg[0];
#pragma unroll
  for (int o = 1; o < 5; ++o) mx = fmaxf(mx, lg[o]);
  float s = 0.0f;
#pragma unroll
  for (int o = 0; o < 5; ++o) { lg[o] = __expf(lg[o] - mx); s += lg[o]; }
  float inv = 1.0f / s;
#pragma unroll
  for (int o = 0; o < 5; ++o) out[(size_t)n * 5 + o] = lg[o] * inv;
}

// ----------------------------- host driver --------------------------------

static inline int cdiv(int a, int b) { return (a + b - 1) / b; }

extern "C" void kernel_launch(void* const* d_in, const int* in_sizes, int n_in,
                              void* d_out, int out_size, void* d_ws, size_t ws_size,
                              hipStream_t stream) {
  const float* x       = (const float*)d_in[0];
  const int*   ei      = (const int*)  d_in[1];
  const float* proj_W  = (const float*)d_in[2];
  const float* proj_b  = (const float*)d_in[3];
  const float* conv1_W = (const float*)d_in[4];
  const float* conv1_b = (const float*)d_in[5];
  const float* conv2_W = (const float*)d_in[6];
  const float* conv2_b = (const float*)d_in[7];
  const float* hc      = (const float*)d_in[8];
  const float* gWih    = (const float*)d_in[9];
  const float* gWhh    = (const float*)d_in[10];
  const float* gbih    = (const float*)d_in[11];
  const float* gbhh    = (const float*)d_in[12];
  const float* out_W   = (const float*)d_in[13];
  const float* out_b   = (const float*)d_in[14];

  const int N = in_sizes[0] / 16;   // 50000
  const int E = in_sizes[1] / 2;    // 800000
  const int NF = in_sizes[0] / N;   // 16

  float* ws   = (float*)d_ws;
  float* dis  = ws;
  size_t off  = ((size_t)N + 63) & ~(size_t)63;
  float* h    = ws + off;                      // N x 128
  float* buf1 = h    + (size_t)N * HID;        // N x 128 (acc / xin)
  float* buf2 = buf1 + (size_t)N * HID;        // N x 128 (h @ W)

  const int*  src = ei;
  const int*  dst = ei + E;
  const int   nb16 = cdiv(N, 16);
  const int   tNH  = N * HID;

  // GCN normalization: deg (with self-loops) -> dis = deg^-1/2
  k_fill1<<<cdiv(N, 256), 256, 0, stream>>>(dis, N);
  k_edge_deg<<<cdiv(E, 256), 256, 0, stream>>>(dis, dst, E);
  k_rsqrt<<<cdiv(N, 256), 256, 0, stream>>>(dis, N);

  // h = broadcast(hidden_c); xin = x @ proj_W + proj_b; h = GRU(xin, h)
  k_init_h<<<cdiv(tNH, 256), 256, 0, stream>>>(h, hc, tNH);
  k_gemm16<<<nb16, 256, 0, stream>>>(x, proj_W, proj_b, buf1, N, NF, 0);
  k_gru<<<nb16, 256, 0, stream>>>(buf1, 0, h, gWih, gWhh, gbih, gbhh, N);

  for (int it = 0; it < 3; ++it) {
    // a = relu(gcn(h, conv1))
    k_gemm16<<<nb16, 256, 0, stream>>>(h, conv1_W, nullptr, buf2, N, HID, 0);
    k_scatter_init<<<cdiv(tNH, 256), 256, 0, stream>>>(buf1, buf2, dis, conv1_b, tNH);
    k_edge_scatter<<<cdiv(E * 32, 256), 256, 0, stream>>>(buf1, buf2, dis, src, dst, E);
    // a = relu(gcn(relu(a), conv2))
    k_gemm16<<<nb16, 256, 0, stream>>>(buf1, conv2_W, nullptr, buf2, N, HID, 1);
    k_scatter_init<<<cdiv(tNH, 256), 256, 0, stream>>>(buf1, buf2, dis, conv2_b, tNH);
    k_edge_scatter<<<cdiv(E * 32, 256), 256, 0, stream>>>(buf1, buf2, dis, src, dst, E);
    // h = GRU(relu(a), h)
    k_gru<<<nb16, 256, 0, stream>>>(buf1, 1, h, gWih, gWhh, gbih, gbhh, N);
  }

  k_out_softmax<<<cdiv(N, 256), 256, 0, stream>>>(h, out_W, out_b, (float*)d_out, N);
}